// ChamferDistance_76811195121928
// MI455X (gfx1250) — compile-verified
//
#include <hip/hip_runtime.h>

typedef __attribute__((ext_vector_type(2))) float v2f;
typedef __attribute__((ext_vector_type(8))) float v8f;

#define MAX_Q 4096   // reference: M = N = 4096

__global__ void chamfer_zero_kernel(float* out) {
    if (threadIdx.x == 0 && blockIdx.x == 0) out[0] = 0.0f;
}

// Block = 8 waves = 8 consecutive 16-row tiles of P, all in the same batch
// (tilesPerB % 8 == 0). The block stages the whole target set Q into LDS in
// B-matrix VGPR layout once; each wave then sweeps all 16-col tiles with
// ds_load_b64 + v_wmma_f32_16x16x4_f32 + v_min.
//
// Math: A row = (-2x1, -2y1, -2z1, 1), B col = (x2, y2, z2, ||q||^2), C = 0
//   =>  D'[i][j] = ||q_j||^2 - 2 p_i.q_j  =  d^2(i,j) - ||p_i||^2
// ||p_i||^2 is row-constant, so it commutes with min over j and is added
// once after the reduction (as a tile-wide sum).
__global__ __launch_bounds__(256)
void chamfer_dir_kernel(const float* __restrict__ P, const float* __restrict__ Q,
                        float* __restrict__ out,
                        int nP, int nQ, int tilesPerB, int qTiles, int totalTiles,
                        float scale)
{
    __shared__ float Bs[MAX_Q * 4];   // 64 KB: per tile 32 lanes x float2

    const int lane = threadIdx.x & 31;
    const int wave = threadIdx.x >> 5;
    const int tile = blockIdx.x * 8 + wave;
    const int b    = (blockIdx.x * 8) / tilesPerB;   // uniform across block
    const int l16  = lane & 15;
    const bool hi  = lane >= 16;

    // ---- Stage Q into LDS in B-layout (cooperative, once per block).
    // Tile jt, lane l (<16): (x,y) of point jt*16+l ; lane 16+l: (z, n2).
    {
        const float* qb = Q + (size_t)b * nQ * 3;
        for (int idx = threadIdx.x; idx < nQ; idx += 256) {
            const float* q = qb + (size_t)idx * 3;
            float x = q[0], y = q[1], z = q[2];
            float n2 = x * x + y * y + z * z;
            int jt = idx >> 4, l = idx & 15;
            float* t = Bs + (size_t)jt * 64;
            t[l * 2 + 0]        = x;
            t[l * 2 + 1]        = y;
            t[32 + l * 2 + 0]   = z;
            t[32 + l * 2 + 1]   = n2;
        }
    }
    __syncthreads();
    if (tile >= totalTiles) return;

    // ---- Build A (16x4 f32, 2 VGPRs). Lanes 0-15: K=0/1, lanes 16-31: K=2/3
    // for the same 16 rows. A row = (-2x, -2y, -2z, 1).
    const int r = tile % tilesPerB * 16 + l16;
    const float* pp = P + (size_t)(b * nP + r) * 3;
    float x1 = pp[0], y1 = pp[1], z1 = pp[2];
    float n1 = x1 * x1 + y1 * y1 + z1 * z1;
    v2f A;
    A.x = hi ? (-2.0f * z1) : (-2.0f * x1);
    A.y = hi ? 1.0f         : (-2.0f * y1);

    v8f Cz = {};          // C = 0, loop invariant
    v8f rm;
#pragma unroll
    for (int v = 0; v < 8; ++v) rm[v] = 3.4e38f;

    // ---- Hot loop: one ds_load_b64 + one WMMA + running min per 16x16 tile.
    const v2f* bs = (const v2f*)(Bs) + lane;        // lane's slot, stride 32
    for (int jt = 0; jt < qTiles; ++jt) {
        v2f Bm = bs[jt * 32];
        v8f D = __builtin_amdgcn_wmma_f32_16x16x4_f32(
            false, A, false, Bm, (short)0, Cz, false, false);
#pragma unroll
        for (int v = 0; v < 8; ++v) rm[v] = fminf(rm[v], D[v]);
    }

    // ---- Reduction. D layout: VGPR v, lanes 0-15 = row v, lanes 16-31 =
    // row v+8, column = lane%16. Butterfly-min (masks 1,2,4,8) stays within
    // each 16-lane half.
    float s = 0.0f;
#pragma unroll
    for (int v = 0; v < 8; ++v) {
        float m = rm[v];
        m = fminf(m, __shfl_xor(m, 1, 32));
        m = fminf(m, __shfl_xor(m, 2, 32));
        m = fminf(m, __shfl_xor(m, 4, 32));
        m = fminf(m, __shfl_xor(m, 8, 32));
        s += m;                       // halves: rows 0-7 | rows 8-15
    }
    s += __shfl_xor(s, 16, 32);       // sum of rowmin(D') over all 16 rows

    // Add back the row-constant ||p||^2: butterfly-add over one half sums
    // n1 of all 16 rows (each half holds the full set).
    float t = n1;
    t += __shfl_xor(t, 1, 32);
    t += __shfl_xor(t, 2, 32);
    t += __shfl_xor(t, 4, 32);
    t += __shfl_xor(t, 8, 32);

    if (lane == 0) atomicAdd(out, (s + t) * scale);
}

extern "C" void kernel_launch(void* const* d_in, const int* in_sizes, int n_in,
                              void* d_out, int out_size, void* d_ws, size_t ws_size,
                              hipStream_t stream) {
    const float* xyz1 = (const float*)d_in[0];
    const float* xyz2 = (const float*)d_in[1];
    float* out = (float*)d_out;

    const int B = 8;                         // per reference
    const int N = in_sizes[0] / (B * 3);     // 4096
    const int M = in_sizes[1] / (B * 3);     // 4096

    chamfer_zero_kernel<<<1, 32, 0, stream>>>(out);

    // Pass 1: dist1 = min over xyz2 for each xyz1 point
    {
        const int tilesPerB = N / 16;
        const int qTiles    = M / 16;
        const int total     = B * tilesPerB;
        const int blocks    = (total + 7) / 8;
        chamfer_dir_kernel<<<blocks, 256, 0, stream>>>(
            xyz1, xyz2, out, N, M, tilesPerB, qTiles, total, 1.0f / (float)(B * N));
    }
    // Pass 2: dist2 = min over xyz1 for each xyz2 point (roles swapped)
    {
        const int tilesPerB = M / 16;
        const int qTiles    = N / 16;
        const int total     = B * tilesPerB;
        const int blocks    = (total + 7) / 8;
        chamfer_dir_kernel<<<blocks, 256, 0, stream>>>(
            xyz2, xyz1, out, M, N, tilesPerB, qTiles, total, 1.0f / (float)(B * M));
    }
}